// NearestNeighborGraph_50560355008512
// MI455X (gfx1250) — compile-verified
//
#include <hip/hip_runtime.h>
#include <hip/hip_bf16.h>
#include <math.h>

typedef float v2f __attribute__((ext_vector_type(2)));
typedef float v8f __attribute__((ext_vector_type(8)));

#define NSETS 4
#define M     4096
#define D     128
#define KNN   16
#define WAVES 4
#define TILES_PER_WAVE (M / 16 / WAVES)   // 64 column tiles per wave

// ---------------------------------------------------------------------------
// Kernel 1: per-point squared norms -> d_ws (N*M floats = 64KB)
// ---------------------------------------------------------------------------
__global__ __launch_bounds__(256)
void knn_norms_kernel(const float* __restrict__ x, float* __restrict__ norms) {
    int p = blockIdx.x * blockDim.x + threadIdx.x;   // 0 .. N*M-1
    const float4* row = (const float4*)(x + (size_t)p * D);
    float s = 0.f;
#pragma unroll
    for (int t = 0; t < D / 4; ++t) {
        float4 v = row[t];
        s += v.x * v.x + v.y * v.y + v.z * v.z + v.w * v.w;
    }
    norms[p] = s;
}

// ---------------------------------------------------------------------------
// Kernel 2: fused WMMA Gram tiles + streaming top-k selection.
// Block = 128 threads (4 waves). blockIdx.x = row tile (0..255), blockIdx.y = set.
// ---------------------------------------------------------------------------
__global__ __launch_bounds__(128)
void knn_graph_kernel(const float* __restrict__ x,
                      const float* __restrict__ norms,
                      int* __restrict__ out_src,
                      int* __restrict__ out_dst) {
    __shared__ float rowsLDS[16 * D];                 // 8 KB  : A tile (shared by all waves)
    __shared__ float colsLDS[WAVES][16 * D];          // 32 KB : per-wave B tile
    __shared__ float tileLDS[WAVES][16 * 16];         // 4 KB  : per-wave distance tile
    __shared__ float mergeD[16][8 * KNN];             // 8 KB  : per-row candidate dists
    __shared__ int   mergeI[16][8 * KNN];             // 8 KB  : per-row candidate indices

    const int n    = blockIdx.y;
    const int r0   = blockIdx.x * 16;
    const int tid  = threadIdx.x;
    const int wave = tid >> 5;
    const int lane = tid & 31;
    const int lane15 = lane & 15;      // M (A) / N (B) index within fragment
    const int khalf  = lane >> 4;      // selects K pair {2h, 2h+1}

    const float* xs = x + (size_t)n * M * D;
    const float* ns = norms + (size_t)n * M;

    // ---- stage the 16 rows (contiguous 8KB) coalesced across the block ----
    {
        const float4* g = (const float4*)(xs + (size_t)r0 * D);
        float4* l = (float4*)rowsLDS;
        for (int t = tid; t < 16 * D / 4; t += 128) l[t] = g[t];
    }
    __syncthreads();

    // ---- A fragments are invariant across all column tiles: hoist into
    //      registers once (32 x v2f = 64 VGPRs/lane) ----
    v2f afrag[D / 4];
#pragma unroll
    for (int k0 = 0; k0 < D; k0 += 4)
        afrag[k0 / 4] = *(const v2f*)&rowsLDS[lane15 * D + k0 + 2 * khalf];

    // ---- per-lane running top-16 (tracked-max replacement) ----
    float bestD[KNN];
    int   bestI[KNN];
#pragma unroll
    for (int t = 0; t < KNN; ++t) { bestD[t] = INFINITY; bestI[t] = 0; }
    float curMax = INFINITY;
    int   maxPos = 0;

    for (int it = 0; it < TILES_PER_WAVE; ++it) {
        const int j0 = (wave + WAVES * it) * 16;      // this wave's column tile

        // prefetch next column tile for this wave (global_prefetch_b8)
        if (it + 1 < TILES_PER_WAVE) {
            const int jn = (wave + WAVES * (it + 1)) * 16;
            __builtin_prefetch(xs + (size_t)jn * D + lane * 16, 0, 1);
        }

        // ---- stage 16 columns (contiguous 8KB) coalesced within the wave ----
        {
            const float4* g = (const float4*)(xs + (size_t)j0 * D);
            float4* l = (float4*)colsLDS[wave];
            for (int t = lane; t < 16 * D / 4; t += 32) l[t] = g[t];
        }
        __syncthreads();

        // ---- 16x16 Gram tile: 32 x V_WMMA_F32_16X16X4_F32 over K=128.
        //      Two accumulators break the serial C-dependency chain. ----
        v8f acc0 = {};
        v8f acc1 = {};
#pragma unroll
        for (int k0 = 0; k0 < D; k0 += 8) {
            const int ks = k0 + 2 * khalf;
            v2f b0 = *(const v2f*)&colsLDS[wave][lane15 * D + ks];
            v2f b1 = *(const v2f*)&colsLDS[wave][lane15 * D + ks + 4];
            acc0 = __builtin_amdgcn_wmma_f32_16x16x4_f32(
                false, afrag[k0 / 4],     false, b0, (short)0, acc0, false, false);
            acc1 = __builtin_amdgcn_wmma_f32_16x16x4_f32(
                false, afrag[k0 / 4 + 1], false, b1, (short)0, acc1, false, false);
        }

        // dist(i,j) ranking key = norm_j - 2*dot  (row norm is a constant shift)
        const float cn = ns[j0 + lane15];             // this lane's column norm
#pragma unroll
        for (int v = 0; v < 8; ++v) {
            const int r = v + 8 * khalf;              // C layout: M = v (+8 for upper half)
            tileLDS[wave][r * 16 + lane15] = fmaf(-2.0f, acc0[v] + acc1[v], cn);
        }
        __syncthreads();

        // ---- scan: lane owns (row = lane&15, cols khalf*8 .. +7) ----
        const int cbase = khalf * 8;
#pragma unroll
        for (int c = 0; c < 8; ++c) {
            const float dch = tileLDS[wave][lane15 * 16 + cbase + c];
            const int   jix = j0 + cbase + c;
            if (dch < curMax) {
                bestD[maxPos] = dch;
                bestI[maxPos] = jix;
                curMax = bestD[0]; maxPos = 0;
#pragma unroll
                for (int t = 1; t < KNN; ++t)
                    if (bestD[t] > curMax) { curMax = bestD[t]; maxPos = t; }
            }
        }
        __syncthreads();
    }

    // ---- dump 8 partial lists per row (4 waves x 2 half-lanes) ----
    {
        const int listid = wave * 2 + khalf;          // 0..7
#pragma unroll
        for (int t = 0; t < KNN; ++t) {
            mergeD[lane15][listid * KNN + t] = bestD[t];
            mergeI[lane15][listid * KNN + t] = bestI[t];
        }
    }
    __syncthreads();

    // ---- final: 16 owner threads pick 16 smallest of 128, ascending ----
    if (tid < 16) {
        const int row  = tid;
        const int myId = n * M + r0 + row;
        const size_t base = (size_t)myId * KNN;
        for (int t = 0; t < KNN; ++t) {
            float mn = INFINITY; int mpos = 0;
            for (int c = 0; c < 8 * KNN; ++c) {
                const float dv = mergeD[row][c];
                if (dv < mn) { mn = dv; mpos = c; }
            }
            out_src[base + t] = n * M + mergeI[row][mpos];
            out_dst[base + t] = myId;
            mergeD[row][mpos] = INFINITY;
        }
    }
}

// ---------------------------------------------------------------------------
extern "C" void kernel_launch(void* const* d_in, const int* in_sizes, int n_in,
                              void* d_out, int out_size, void* d_ws, size_t ws_size,
                              hipStream_t stream) {
    const float* x   = (const float*)d_in[0];
    float* norms     = (float*)d_ws;                  // N*M floats = 64KB
    int*   out       = (int*)d_out;
    int*   out_src   = out;                           // first  N*M*K int32
    int*   out_dst   = out + (size_t)NSETS * M * KNN; // second N*M*K int32

    knn_norms_kernel<<<(NSETS * M) / 256, 256, 0, stream>>>(x, norms);
    knn_graph_kernel<<<dim3(M / 16, NSETS), 128, 0, stream>>>(x, norms, out_src, out_dst);
}